// SVF_2D_34900904247315
// MI455X (gfx1250) — compile-verified
//
#include <hip/hip_runtime.h>

// SVF scaling-and-squaring, 2D, N=8, H=W=1024, 12 steps.
// Memory-bound but L2-resident (128 MB ping-pong < 192 MB MI455X L2), so the
// optimization target is VMEM issue count + latency overlap, not HBM bytes.
// CDNA5 paths used: GLOBAL_LOAD_ASYNC_TO_LDS_B32 (ASYNCcnt) + s_wait_asynccnt.

namespace {
constexpr int kW = 1024;
constexpr int kH = 1024;
constexpr int kNB = 8;
constexpr int kSteps = 12;
constexpr int kHW = kW * kH;        // 1 << 20
constexpr int kNPix = kNB * kHW;    // 1 << 23
constexpr float kSX = 2.0f / (kW - 1);
constexpr float kSY = 2.0f / (kH - 1);
}

// Async copy of one 32-bit element per lane: global -> LDS (VDST = LDS byte
// offset, VADDR = 64-bit global address). Tracked by ASYNCcnt.
__device__ __forceinline__ void async_g2l_b32(const float* g, const float* lds_slot) {
  unsigned lds_off = (unsigned)(unsigned long long)lds_slot;  // low 32 bits of flat LDS addr
  asm volatile("global_load_async_to_lds_b32 %0, %1, off"
               :
               : "v"(lds_off), "v"((unsigned long long)g)
               : "memory");
}

__device__ __forceinline__ void wait_async0() {
#if __has_builtin(__builtin_amdgcn_s_wait_asynccnt)
  __builtin_amdgcn_s_wait_asynccnt(0);
#else
  asm volatile("s_wait_asynccnt 0x0" ::: "memory");
#endif
  asm volatile("" ::: "memory");  // keep LDS reads below the wait
}

// Unaligned-capable 2-float global load (gfx1250 supports unaligned b64).
__device__ __forceinline__ float2 ldg2(const float* p) {
  float2 r;
  __builtin_memcpy(&r, p, sizeof(float2));
  return r;
}

// One squaring step. FUSE_INIT=true reads the raw velocity field v and applies
// disp0 = v*scale/4096 implicitly (scaling commutes with bilinear sampling):
//   px = x + v_x/4096,  out = (scale/4096) * (v_center + bilin(v)).
template <bool FUSE_INIT>
__global__ __launch_bounds__(256) void svf_step_t(const float* __restrict__ in,
                                                  float* __restrict__ out) {
  __shared__ float sh[512];
  const int tid  = threadIdx.x;
  const int i    = blockIdx.x * 256 + tid;        // pixel id in [0, kNPix)
  const int rem  = i & (kHW - 1);
  const int base = (i >> 20) << 21;               // n * 2 * kHW
  const int x    = rem & (kW - 1);
  const int y    = rem >> 10;

  const float* cx = in + base;                    // x-displacement (or v_x) plane
  const float* cy = cx + kHW;                     // y-displacement (or v_y) plane

  // CDNA5 async copy path for the streaming center values.
  async_g2l_b32(cx + rem, &sh[tid]);
  async_g2l_b32(cy + rem, &sh[256 + tid]);
  wait_async0();
  const float ddx = sh[tid];
  const float ddy = sh[256 + tid];

  // Sample position in pixels (border clamp, align_corners).
  float px, py;
  if (FUSE_INIT) {
    px = fmaf(ddx, 1.0f / 4096.0f, (float)x);     // (scale/4096)*(W-1)/2 == 1/4096
    py = fmaf(ddy, 1.0f / 4096.0f, (float)y);
  } else {
    px = fmaf(ddx, 0.5f * (kW - 1), (float)x);
    py = fmaf(ddy, 0.5f * (kH - 1), (float)y);
  }
  px = fminf(fmaxf(px, 0.0f), (float)(kW - 1));
  py = fminf(fmaxf(py, 0.0f), (float)(kH - 1));

  const float fx0 = floorf(px);
  const float fy0 = floorf(py);
  const float wx = px - fx0;
  const float wy = py - fy0;
  const int x0 = (int)fx0;
  const int y0 = (int)fy0;
  const int y1 = min(y0 + 1, kH - 1);

  // Horizontal tap pair as ONE b64 gather: x1 == x0+1 except when x0 == W-1,
  // where the clamp forces wx == 0 so only v00 matters. Load at xb=min(x0,W-2)
  // (always in-bounds) and select. 4 b64 gathers replace 8 b32 gathers.
  const int  xb = min(x0, kW - 2);
  const bool xe = (x0 == kW - 1);
  const int  r0 = y0 * kW + xb;
  const int  r1 = y1 * kW + xb;

  const float2 a0 = ldg2(cx + r0);
  const float2 a1 = ldg2(cx + r1);
  const float2 b0 = ldg2(cy + r0);
  const float2 b1 = ldg2(cy + r1);

  const float a00 = xe ? a0.y : a0.x, a01 = a0.y;
  const float a10 = xe ? a1.y : a1.x, a11 = a1.y;
  const float b00 = xe ? b0.y : b0.x, b01 = b0.y;
  const float b10 = xe ? b1.y : b1.x, b11 = b1.y;

  const float at = fmaf(wx, a01 - a00, a00);
  const float ab = fmaf(wx, a11 - a10, a10);
  const float sa = fmaf(wy, ab - at, at);
  const float bt = fmaf(wx, b01 - b00, b00);
  const float bb = fmaf(wx, b11 - b10, b10);
  const float sb = fmaf(wy, bb - bt, bt);

  if (FUSE_INIT) {
    out[base + rem]       = (kSX / 4096.0f) * (ddx + sa);
    out[base + kHW + rem] = (kSY / 4096.0f) * (ddy + sb);
  } else {
    out[base + rem]       = ddx + sa;
    out[base + kHW + rem] = ddy + sb;
  }
}

// Element-wise epilogue: trans = identity + disp, disp_vox = disp/scale.
// Reads disp from `dvox` buffer and rewrites it in place (per-element safe).
__global__ __launch_bounds__(256) void svf_final(float* __restrict__ trans,
                                                 float* __restrict__ dvox) {
  const int i    = blockIdx.x * 256 + threadIdx.x;
  const int rem  = i & (kHW - 1);
  const int base = (i >> 20) << 21;
  const int x    = rem & (kW - 1);
  const int y    = rem >> 10;

  const float dx = dvox[base + rem];
  const float dy = dvox[base + kHW + rem];

  trans[base + rem]       = fmaf((float)x, kSX, -1.0f) + dx;
  trans[base + kHW + rem] = fmaf((float)y, kSY, -1.0f) + dy;
  dvox[base + rem]        = dx * ((kW - 1) * 0.5f);
  dvox[base + kHW + rem]  = dy * ((kH - 1) * 0.5f);
}

extern "C" void kernel_launch(void* const* d_in, const int* in_sizes, int n_in,
                              void* d_out, int out_size, void* d_ws, size_t ws_size,
                              hipStream_t stream) {
  (void)in_sizes; (void)n_in; (void)out_size; (void)d_ws; (void)ws_size;
  const float* v = (const float*)d_in[0];
  // d_in[1] (identity_grid) is analytic; recomputed on the fly.

  float* bufA = (float*)d_out;                       // first half: `transformation`
  float* bufB = bufA + (size_t)2 * kNB * kHW;        // second half: `disp_vox`

  const int blocks = kNPix / 256;                    // 32768 blocks of 256 (8 waves)

  // Step 1 fused with init: v -> bufA.
  svf_step_t<true><<<blocks, 256, 0, stream>>>(v, bufA);

  // Steps 2..12 (11 steps): A->B, B->A, ..., 11th writes bufB.
  float* src = bufA;
  float* dst = bufB;
  for (int s = 0; s < kSteps - 1; ++s) {
    svf_step_t<false><<<blocks, 256, 0, stream>>>(src, dst);
    float* t = src; src = dst; dst = t;
  }
  // Final disp resides in bufB; convert to (transformation, disp_vox).
  svf_final<<<blocks, 256, 0, stream>>>(bufA, bufB);
}